// DeepSeekV2Layer_40750649704540
// MI455X (gfx1250) — compile-verified
//
#include <hip/hip_runtime.h>
#include <hip/hip_bf16.h>
#include <math.h>

#define Bsz 64
#define Ssz 4096
#define HID 4096
#define NH 32
#define HD 128
#define QSIZE (NH*HD)         // 4096
#define KVSIZE 576            // 512 + 64
#define QKVN (QSIZE + KVSIZE) // 4672
#define NEXP 8
#define MI 1408
#define GUN (2*MI)            // 2816

typedef __attribute__((ext_vector_type(2))) float v2f;
typedef __attribute__((ext_vector_type(8))) float v8f;
typedef __attribute__((ext_vector_type(4))) unsigned int u32x4;
typedef __attribute__((ext_vector_type(8))) int i32x8;
typedef __attribute__((ext_vector_type(4))) int i32x4;

#define BK    64              // K-chunk per TDM tile
#define TROWS 16              // weight rows per wave tile
#define LDSW  (BK + 2)        // row stride in LDS after TDM pad (2 DWORDs/row)

#if __has_builtin(__builtin_amdgcn_tensor_load_to_lds)
#define HAVE_TDM 1
#else
#define HAVE_TDM 0
#endif

#if HAVE_TDM
// ---------------------------------------------------------------------------
// Issue one TDM load: 2-D tile, TROWS rows x BK f32, row stride K elements,
// into LDS at byte offset lds_byte with 2-DWORD padding after each 64-DWORD row.
// Descriptor layout per CDNA5 ISA.
// ---------------------------------------------------------------------------
__device__ __forceinline__ void tdm_issue(const float* gaddr, unsigned lds_byte, int K)
{
    unsigned long long ga = (unsigned long long)(uintptr_t)gaddr;
    u32x4 g0;
    g0[0] = 1u;                                        // count=1 (valid), no gather
    g0[1] = lds_byte;                                  // lds_addr (bytes)
    g0[2] = (unsigned)(ga & 0xFFFFFFFFu);              // global_addr[31:0]
    g0[3] = (unsigned)((ga >> 32) & 0x01FFFFFFu)       // global_addr[56:32]
          | 0x80000000u;                               // type=2 ("image")
    i32x8 g1;
    g1[0] = (2 << 16)        // data_size = 4 bytes
          | (1 << 20)        // pad_enable
          | (5 << 22)        // pad_interval: every 64 DWORDs (one row)
          | (1 << 25);       // pad_amount: 2 DWORDs
    g1[1] = (BK & 0xFFFF) << 16;       // tensor_dim0[15:0] = BK
    g1[2] = (TROWS & 0xFFFF) << 16;    // tensor_dim0 hi=0 | tensor_dim1[15:0]=TROWS
    g1[3] = (BK << 16);                // tensor_dim1 hi=0 | tile_dim0 = BK
    g1[4] = TROWS;                     // tile_dim1 = TROWS, tile_dim2 = 0
    g1[5] = K;                         // tensor_dim0_stride[31:0] (elements)
    g1[6] = 0;                         // stride hi / dim1_stride lo
    g1[7] = 0;
    i32x4 z4 = {0, 0, 0, 0};
#if defined(__clang_major__) && __clang_major__ >= 23
    i32x8 z8 = {0, 0, 0, 0, 0, 0, 0, 0};
    __builtin_amdgcn_tensor_load_to_lds(g0, g1, z4, z4, z8, 0);
#else
    __builtin_amdgcn_tensor_load_to_lds(g0, g1, z4, z4, 0);
#endif
}
#endif

// ---------------------------------------------------------------------------
// Generic skinny GEMM:  C[M x N] = A[M x K] * W[N x K]^T  (row-major W)
// Wave-level WMMA f32 16x16x4.  4 waves/block, each wave -> one 16x16 tile.
// Weight rows are streamed into LDS by the Tensor Data Mover (double-buffered,
// wave-private tiles -> no cross-wave barriers; sync via s_wait_tensorcnt).
// grid = (N/64, M/16).  K must be a multiple of 64.
// ---------------------------------------------------------------------------
__global__ __launch_bounds__(128) void gemm_wmma_f32(
    const float* __restrict__ A, const float* __restrict__ W,
    float* __restrict__ C, const float* __restrict__ residual,
    const float* __restrict__ rowscale, int M, int N, int K, int accumulate)
{
    const int wave = threadIdx.x >> 5;
    const int lane = threadIdx.x & 31;
    const int h    = lane >> 4;   // half-wave select
    const int p    = lane & 15;   // 0..15
    const int m0   = blockIdx.y * 16;
    const int n0   = blockIdx.x * 64 + wave * 16;

    const float* __restrict__ Arow = A + (size_t)(m0 + p) * K;   // A frag: M = p
    v8f acc = {};

#if HAVE_TDM
    __shared__ float sW[4 * 2 * TROWS * LDSW];
    const int wbase = wave * (2 * TROWS * LDSW);
    unsigned lds0 = (unsigned)(uintptr_t)&sW[wbase];
    unsigned lds1 = (unsigned)(uintptr_t)&sW[wbase + TROWS * LDSW];
    const float* gW = W + (size_t)n0 * K;   // this wave's 16 weight rows

    const int nch = K / BK;
    tdm_issue(gW, lds0, K);                 // prime buffer 0
    for (int c = 0; c < nch; ++c) {
        if (c + 1 < nch) {
            tdm_issue(gW + (size_t)(c + 1) * BK, (c & 1) ? lds0 : lds1, K);
            __builtin_amdgcn_s_wait_tensorcnt(1);   // oldest (current) tile done
        } else {
            __builtin_amdgcn_s_wait_tensorcnt(0);
        }
        const int bb = wbase + ((c & 1) ? TROWS * LDSW : 0) + p * LDSW;
        #pragma unroll
        for (int kk = 0; kk < BK; kk += 4) {
            v2f b = *(const v2f*)&sW[bb + kk + 2*h];           // ds_load_b64
            v2f a = *(const v2f*)(Arow + c * BK + kk + 2*h);   // L2-hot A
            acc = __builtin_amdgcn_wmma_f32_16x16x4_f32(
                false, a, false, b, (short)0, acc, false, false);
        }
    }
#else
    const float* __restrict__ Wrow = W + (size_t)(n0 + p) * K;
    for (int k = 0; k < K; k += 64) {
        #pragma unroll
        for (int kk = 0; kk < 64; kk += 4) {
            v2f a = *(const v2f*)(Arow + k + kk + 2*h);
            v2f b = *(const v2f*)(Wrow + k + kk + 2*h);
            acc = __builtin_amdgcn_wmma_f32_16x16x4_f32(
                false, a, false, b, (short)0, acc, false, false);
        }
    }
#endif

    // C/D layout: lane l -> N = l%16 ; VGPR v -> M = v + 8*(l/16)
    #pragma unroll
    for (int v = 0; v < 8; ++v) {
        int m = m0 + v + 8*h;
        int n = n0 + p;
        size_t idx = (size_t)m * N + n;
        float val = acc[v];
        if (rowscale)  val *= rowscale[m];
        if (residual)  val += residual[idx];
        if (accumulate) C[idx] += val; else C[idx] = val;
    }
}

// ---------------------------------------------------------------------------
// RMSNorm: one block per row of HID
// ---------------------------------------------------------------------------
__global__ __launch_bounds__(256) void rmsnorm_kernel(
    const float* __restrict__ x, const float* __restrict__ w,
    float* __restrict__ out, int n)
{
    const int b = blockIdx.x;
    const int lane = threadIdx.x & 31, wave = threadIdx.x >> 5;
    const float* xr = x + (size_t)b * n;
    float ss = 0.f;
    for (int i = threadIdx.x; i < n; i += blockDim.x) { float v = xr[i]; ss += v*v; }
    #pragma unroll
    for (int off = 16; off; off >>= 1) ss += __shfl_xor(ss, off, 32);
    __shared__ float red[9];
    if (lane == 0) red[wave] = ss;
    __syncthreads();
    if (threadIdx.x == 0) {
        float s = 0.f;
        for (int i = 0; i < 8; ++i) s += red[i];
        red[8] = rsqrtf(s / n + 1e-6f);
    }
    __syncthreads();
    float r = red[8];
    for (int i = threadIdx.x; i < n; i += blockDim.x)
        out[(size_t)b * n + i] = xr[i] * r * w[i];
}

// ---------------------------------------------------------------------------
// Flash-decode MQA attention.  grid = (NH, B), 256 threads = 8 waves.
// Each wave handles positions s = wave, wave+8, ... with online softmax.
// Position len-1 uses kcur (for BOTH k and v, per the reference).
// ---------------------------------------------------------------------------
__global__ __launch_bounds__(256) void attn_kernel(
    const float* __restrict__ qkv, const float* __restrict__ k_cache,
    const float* __restrict__ v_cache, const int* __restrict__ seq_lens,
    float* __restrict__ ctx_out)
{
    const int hd = blockIdx.x, b = blockIdx.y;
    const int lane = threadIdx.x & 31, wave = threadIdx.x >> 5;
    const int len = seq_lens[b];

    const float* q    = qkv + (size_t)b * QKVN + hd * HD;
    const float* kcur = qkv + (size_t)b * QKVN + QSIZE;
    const float scale = rsqrtf((float)HD);

    float4 qv = *(const float4*)(q + lane * 4);

    float m = -INFINITY, l = 0.f;
    float ax = 0.f, ay = 0.f, az = 0.f, aw = 0.f;

    for (int s = wave; s < len; s += 8) {
        const float* krow = (s == len - 1) ? kcur : (k_cache + ((size_t)b * Ssz + s) * HD);
        const float* vrow = (s == len - 1) ? kcur : (v_cache + ((size_t)b * Ssz + s) * HD);
        float4 kv = *(const float4*)(krow + lane * 4);
        float dot = qv.x*kv.x + qv.y*kv.y + qv.z*kv.z + qv.w*kv.w;
        #pragma unroll
        for (int off = 16; off; off >>= 1) dot += __shfl_xor(dot, off, 32);
        float sc   = dot * scale;
        float mnew = fmaxf(m, sc);
        float corr = __expf(m - mnew);
        float pr   = __expf(sc - mnew);
        float4 vv  = *(const float4*)(vrow + lane * 4);
        l  = l * corr + pr;
        ax = ax * corr + pr * vv.x;
        ay = ay * corr + pr * vv.y;
        az = az * corr + pr * vv.z;
        aw = aw * corr + pr * vv.w;
        m  = mnew;
    }

    __shared__ float sm[8], sl[8];
    __shared__ float sacc[8][HD];
    sacc[wave][lane*4+0] = ax; sacc[wave][lane*4+1] = ay;
    sacc[wave][lane*4+2] = az; sacc[wave][lane*4+3] = aw;
    if (lane == 0) { sm[wave] = m; sl[wave] = l; }
    __syncthreads();

    if (wave == 0) {
        float M = -INFINITY;
        #pragma unroll
        for (int w = 0; w < 8; ++w) M = fmaxf(M, sm[w]);
        float L = 0.f, tx = 0.f, ty = 0.f, tz = 0.f, tw = 0.f;
        #pragma unroll
        for (int w = 0; w < 8; ++w) {
            float c = __expf(sm[w] - M);
            L  += sl[w] * c;
            tx += c * sacc[w][lane*4+0];
            ty += c * sacc[w][lane*4+1];
            tz += c * sacc[w][lane*4+2];
            tw += c * sacc[w][lane*4+3];
        }
        float inv = 1.f / L;
        float* o = ctx_out + (size_t)b * QSIZE + hd * HD + lane * 4;
        o[0] = tx*inv; o[1] = ty*inv; o[2] = tz*inv; o[3] = tw*inv;
    }
}

// ---------------------------------------------------------------------------
// Gate: logits = x @ gate_w^T, softmax, top-2, renormalize.
// Writes transposed weights wtsT[e*B + b].
// ---------------------------------------------------------------------------
__global__ __launch_bounds__(256) void gate_kernel(
    const float* __restrict__ x, const float* __restrict__ gw,
    float* __restrict__ wtsT)
{
    const int b = blockIdx.x;
    const int lane = threadIdx.x & 31, wave = threadIdx.x >> 5;
    const float* xr = x + (size_t)b * HID;
    const float* gr = gw + (size_t)wave * HID;
    float dot = 0.f;
    for (int k = lane; k < HID; k += 32) dot += xr[k] * gr[k];
    #pragma unroll
    for (int off = 16; off; off >>= 1) dot += __shfl_xor(dot, off, 32);
    __shared__ float logits[NEXP];
    if (lane == 0) logits[wave] = dot;
    __syncthreads();
    if (threadIdx.x == 0) {
        float mx = -INFINITY;
        for (int e = 0; e < NEXP; ++e) mx = fmaxf(mx, logits[e]);
        float p[NEXP], s = 0.f;
        for (int e = 0; e < NEXP; ++e) { p[e] = __expf(logits[e] - mx); s += p[e]; }
        for (int e = 0; e < NEXP; ++e) p[e] /= s;
        int i0 = 0;
        for (int e = 1; e < NEXP; ++e) if (p[e] > p[i0]) i0 = e;
        int i1 = -1;
        for (int e = 0; e < NEXP; ++e)
            if (e != i0 && (i1 < 0 || p[e] > p[i1])) i1 = e;
        float sum = p[i0] + p[i1];
        for (int e = 0; e < NEXP; ++e)
            wtsT[e * Bsz + b] = (e == i0) ? p[i0]/sum : (e == i1) ? p[i1]/sum : 0.f;
    }
}

// ---------------------------------------------------------------------------
// SiLU(g) * u over the MoE intermediate
// ---------------------------------------------------------------------------
__global__ __launch_bounds__(256) void silu_mul_kernel(
    const float* __restrict__ gu, float* __restrict__ mid)
{
    int idx = blockIdx.x * blockDim.x + threadIdx.x;
    if (idx >= Bsz * MI) return;
    int b = idx / MI, i = idx % MI;
    float g = gu[(size_t)b * GUN + i];
    float u = gu[(size_t)b * GUN + MI + i];
    mid[idx] = (g / (1.f + __expf(-g))) * u;
}

// ---------------------------------------------------------------------------
extern "C" void kernel_launch(void* const* d_in, const int* in_sizes, int n_in,
                              void* d_out, int out_size, void* d_ws, size_t ws_size,
                              hipStream_t stream) {
    (void)in_sizes; (void)n_in; (void)out_size; (void)ws_size;
    const float* hidden  = (const float*)d_in[0];
    const float* k_cache = (const float*)d_in[2];
    const float* v_cache = (const float*)d_in[3];
    const int*   seqlens = (const int*)  d_in[4];
    const float* norm1w  = (const float*)d_in[5];
    const float* norm2w  = (const float*)d_in[6];
    const float* Wqkv    = (const float*)d_in[7];
    const float* Wo      = (const float*)d_in[8];
    const float* gate_w  = (const float*)d_in[9];
    const float* w1      = (const float*)d_in[10];
    const float* w2      = (const float*)d_in[11];
    float* out = (float*)d_out;

    float* ws   = (float*)d_ws;
    float* h1   = ws;                       // B*HID
    float* qkv  = h1   + (size_t)Bsz*HID;   // B*QKVN
    float* ctx  = qkv  + (size_t)Bsz*QKVN;  // B*QSIZE
    float* hid  = ctx  + (size_t)Bsz*QSIZE; // B*HID  (residual + attn_out)
    float* x2   = hid  + (size_t)Bsz*HID;   // B*HID  (rmsnorm2)
    float* wtsT = x2   + (size_t)Bsz*HID;   // E*B
    float* gu   = wtsT + (size_t)NEXP*Bsz;  // B*GUN
    float* mid  = gu   + (size_t)Bsz*GUN;   // B*MI

    // 1) h = rmsnorm(hidden, norm1_w)
    rmsnorm_kernel<<<Bsz, 256, 0, stream>>>(hidden, norm1w, h1, HID);

    // 2) qkv = h @ Wqkv^T   (64 x 4672)
    gemm_wmma_f32<<<dim3(QKVN/64, Bsz/16), 128, 0, stream>>>(
        h1, Wqkv, qkv, nullptr, nullptr, Bsz, QKVN, HID, 0);

    // 3) flash-decode attention -> ctx (64 x 4096)
    attn_kernel<<<dim3(NH, Bsz), 256, 0, stream>>>(qkv, k_cache, v_cache, seqlens, ctx);

    // 4) hid = hidden + ctx @ Wo^T
    gemm_wmma_f32<<<dim3(HID/64, Bsz/16), 128, 0, stream>>>(
        ctx, Wo, hid, hidden, nullptr, Bsz, HID, QSIZE, 0);

    // 5) x2 = rmsnorm(hid, norm2_w)
    rmsnorm_kernel<<<Bsz, 256, 0, stream>>>(hid, norm2w, x2, HID);

    // 6) gate -> top-2 renormalized weights (transposed E x B)
    gate_kernel<<<Bsz, 256, 0, stream>>>(x2, gate_w, wtsT);

    // 7) out = hid  (then accumulate weighted expert outputs)
    hipMemcpyAsync(out, hid, (size_t)Bsz*HID*sizeof(float),
                   hipMemcpyDeviceToDevice, stream);

    // 8) experts: out += wts[:,e] * (silu(g)*u) @ w2[e]^T
    for (int e = 0; e < NEXP; ++e) {
        const float* w1e = w1 + (size_t)e * GUN * HID;
        const float* w2e = w2 + (size_t)e * HID * MI;
        gemm_wmma_f32<<<dim3(GUN/64, Bsz/16), 128, 0, stream>>>(
            x2, w1e, gu, nullptr, nullptr, Bsz, GUN, HID, 0);
        silu_mul_kernel<<<(Bsz*MI + 255)/256, 256, 0, stream>>>(gu, mid);
        gemm_wmma_f32<<<dim3(HID/64, Bsz/16), 128, 0, stream>>>(
            mid, w2e, out, nullptr, wtsT + (size_t)e*Bsz, Bsz, HID, MI, 1);
    }
}